// Universal_7885559956094
// MI455X (gfx1250) — compile-verified
//
#include <hip/hip_runtime.h>

#define FEATS   32
#define CLASSES 16
#define EMBD    6
#define HID     9
#define GDEPTH  10
#define DIFF_C  0.9f

typedef __attribute__((ext_vector_type(2))) float v2f;
typedef __attribute__((ext_vector_type(8))) float v8f;

// ---------------------------------------------------------------- utilities
__global__ void fill_zero_k(float* __restrict__ p, long long n) {
    long long i = (long long)blockIdx.x * blockDim.x + threadIdx.x;
    if (i < n) p[i] = 0.0f;
}

// Narrow int64 edges -> int32 (halves HBM stream for the 20 conv passes)
// and accumulate in-degree in the same pass.
__global__ void edges_prep_k(const long long* __restrict__ e64,
                             int* __restrict__ s32, int* __restrict__ d32,
                             float* __restrict__ deg, long long E) {
    long long i = (long long)blockIdx.x * blockDim.x + threadIdx.x;
    if (i >= E) return;
    int s = (int)e64[i];
    int d = (int)e64[E + i];
    s32[i] = s;
    d32[i] = d;
    atomicAdd(&deg[d], 1.0f);
}

__global__ void norm_k(const int* __restrict__ s32, const int* __restrict__ d32,
                       const float* __restrict__ deg, float* __restrict__ nrm,
                       long long E) {
    long long i = (long long)blockIdx.x * blockDim.x + threadIdx.x;
    if (i >= E) return;
    float ds = fmaxf(deg[s32[i]], 1.0f);
    float dd = fmaxf(deg[d32[i]], 1.0f);
    nrm[i] = rsqrtf(ds * dd);
}

__global__ void init_h_k(const float* __restrict__ x, float* __restrict__ h,
                         float* __restrict__ h0, long long n) {
    long long i = (long long)blockIdx.x * blockDim.x + threadIdx.x;
    if (i < n) { float v = x[i]; h[i] = v; h0[i] = v; }
}

// One thread per (edge, feature): 32 consecutive lanes share one edge, so the
// h[src] gather and acc[dst] scatter are fully coalesced 128B transactions.
// Feature buffers (12.8 MB each) are L2-resident on MI455X (192 MB L2).
__global__ void conv_scatter_k(const float* __restrict__ h,
                               const int* __restrict__ s32, const int* __restrict__ d32,
                               const float* __restrict__ nrm,
                               float* __restrict__ acc, long long E) {
    long long t = (long long)blockIdx.x * blockDim.x + threadIdx.x;
    long long e = t >> 5;
    int f = (int)(t & 31);
    if (e >= E) return;
    int s = s32[e];
    int d = d32[e];
    float w = nrm[e];
    float v = h[(long long)s * FEATS + f] * w;
    atomicAdd(&acc[(long long)d * FEATS + f], v);
}

__global__ void combine_k(const float* __restrict__ acc, const float* __restrict__ h0,
                          float* __restrict__ h, long long n) {
    long long i = (long long)blockIdx.x * blockDim.x + threadIdx.x;
    if (i < n) h[i] = DIFF_C * acc[i] + (1.0f - DIFF_C) * h0[i];
}

// c[f][j] = b1[j] + sum_k emb[f][k] * W1[1+k][j]   (32 x 9 table)
__global__ void mlp_const_k(const float* __restrict__ emb, const float* __restrict__ W1,
                            const float* __restrict__ b1, float* __restrict__ cfj) {
    int t = threadIdx.x;
    if (t >= FEATS * HID) return;
    int f = t / HID, j = t % HID;
    float s = b1[j];
#pragma unroll
    for (int k = 0; k < EMBD; ++k) s += emb[f * EMBD + k] * W1[(1 + k) * HID + j];
    cfj[t] = s;
}

// Collapsed per-(node,feature) MLP: out = b2 + sum_j W2[j]*relu(x*W1[0,j]+c[f,j])
__global__ void mlp_apply_k(float* __restrict__ h, float* __restrict__ h0,
                            const float* __restrict__ W1, const float* __restrict__ W2,
                            const float* __restrict__ b2, const float* __restrict__ cfj,
                            long long n) {
    long long i = (long long)blockIdx.x * blockDim.x + threadIdx.x;
    if (i >= n) return;
    int f = (int)(i & (FEATS - 1));
    float v = h[i];
    float o = b2[0];
#pragma unroll
    for (int j = 0; j < HID; ++j) {
        float t = fmaf(v, W1[j], cfj[f * HID + j]);   // W1[0*HID+j]
        t = fmaxf(t, 0.0f);
        o = fmaf(t, W2[j], o);
    }
    h[i] = o;
    h0[i] = o;
}

// Output GEMM: [N,32] x [32,16] + bout via V_WMMA_F32_16X16X4_F32.
// One wave per 16-row tile, 8 chained f32 WMMAs over K=32. N % 16 == 0 so
// active waves always run with full EXEC (WMMA requirement).
__global__ void out_gemm_wmma_k(const float* __restrict__ h,
                                const float* __restrict__ Wout,
                                const float* __restrict__ bout,
                                float* __restrict__ out, int n) {
    int gtid = blockIdx.x * blockDim.x + threadIdx.x;
    int wave = gtid >> 5;
    int lane = threadIdx.x & 31;
    int row0 = wave * 16;
    if (row0 >= n) return;                    // wave-uniform exit

    int r    = row0 + (lane & 15);            // A: row per lane
    int c    = lane & 15;                     // B/D: col per lane
    int koff = (lane < 16) ? 0 : 2;           // K split across half-waves

    v8f acc = {};
#pragma unroll
    for (int kb = 0; kb < FEATS; kb += 4) {
        int k0 = kb + koff;
        v2f a, b;
        a.x = h[(long long)r * FEATS + k0];
        a.y = h[(long long)r * FEATS + k0 + 1];
        b.x = Wout[(k0    ) * CLASSES + c];
        b.y = Wout[(k0 + 1) * CLASSES + c];
        acc = __builtin_amdgcn_wmma_f32_16x16x4_f32(
            /*neg_a=*/false, a, /*neg_b=*/false, b,
            /*c_mod=*/(short)0, acc, /*reuse_a=*/false, /*reuse_b=*/false);
    }

    float bias = bout[c];
    int rbase = row0 + ((lane < 16) ? 0 : 8); // D: VGPR v -> row v (+8 for hi lanes)
#pragma unroll
    for (int v = 0; v < 8; ++v)
        out[(long long)(rbase + v) * CLASSES + c] = acc[v] + bias;
}

extern "C" void kernel_launch(void* const* d_in, const int* in_sizes, int n_in,
                              void* d_out, int out_size, void* d_ws, size_t ws_size,
                              hipStream_t stream) {
    (void)n_in; (void)out_size; (void)ws_size;

    const float*     x     = (const float*)d_in[0];
    const long long* edges = (const long long*)d_in[1];
    const float*     emb   = (const float*)d_in[2];
    const float*     W1    = (const float*)d_in[3];
    const float*     b1    = (const float*)d_in[4];
    const float*     W2    = (const float*)d_in[5];
    const float*     b2    = (const float*)d_in[6];
    const float*     Wout  = (const float*)d_in[7];
    const float*     bout  = (const float*)d_in[8];
    float*           out   = (float*)d_out;

    const long long Nn = in_sizes[0] / FEATS;
    const long long E  = in_sizes[1] / 2;
    const long long NF = Nn * FEATS;

    // Workspace carve-up (all 4-byte aligned):
    float* deg = (float*)d_ws;              // N
    int*   s32 = (int*)(deg + Nn);          // E
    int*   d32 = s32 + E;                   // E
    float* nrm = (float*)(d32 + E);         // E
    float* h   = nrm + E;                   // N*F
    float* h0  = h + NF;                    // N*F
    float* acc = h0 + NF;                   // N*F
    float* cfj = acc + NF;                  // 32*9

    const int B = 256;
    auto nb = [](long long n, int b) { return (unsigned)((n + b - 1) / b); };

    // degree + edge narrowing + norm (once)
    fill_zero_k<<<nb(Nn, B), B, 0, stream>>>(deg, Nn);
    edges_prep_k<<<nb(E, B), B, 0, stream>>>(edges, s32, d32, deg, E);
    norm_k<<<nb(E, B), B, 0, stream>>>(s32, d32, deg, nrm, E);
    init_h_k<<<nb(NF, B), B, 0, stream>>>(x, h, h0, NF);

    // first diffusion
    for (int it = 0; it < GDEPTH; ++it) {
        fill_zero_k<<<nb(NF, B), B, 0, stream>>>(acc, NF);
        conv_scatter_k<<<nb(E * 32, B), B, 0, stream>>>(h, s32, d32, nrm, acc, E);
        combine_k<<<nb(NF, B), B, 0, stream>>>(acc, h0, h, NF);
    }

    // per-(node,feature) MLP, collapsed
    mlp_const_k<<<1, 288, 0, stream>>>(emb, W1, b1, cfj);
    mlp_apply_k<<<nb(NF, B), B, 0, stream>>>(h, h0, W1, W2, b2, cfj, NF);

    // second diffusion
    for (int it = 0; it < GDEPTH; ++it) {
        fill_zero_k<<<nb(NF, B), B, 0, stream>>>(acc, NF);
        conv_scatter_k<<<nb(E * 32, B), B, 0, stream>>>(h, s32, d32, nrm, acc, E);
        combine_k<<<nb(NF, B), B, 0, stream>>>(acc, h0, h, NF);
    }

    // output GEMM on the WMMA pipe
    long long tiles = (Nn + 15) / 16;
    out_gemm_wmma_k<<<nb(tiles * 32, B), B, 0, stream>>>(h, Wout, bout, out, (int)Nn);
}